// QLSTM_65481071400861
// MI455X (gfx1250) — compile-verified
//
#include <hip/hip_runtime.h>
#include <math.h>

typedef float v2f __attribute__((ext_vector_type(2)));
typedef float v8f __attribute__((ext_vector_type(8)));

#define SEQ 128
#define BATCH 512
#define IN_DIM 128
#define HID 128
#define NQ 8
#define TILE_B 16
#define THREADS 256

__device__ __forceinline__ void lds_wave_sync() {
  // same-wave LDS ordering barrier (CDNA5 split counters)
  asm volatile("s_wait_dscnt 0" ::: "memory");
}

__device__ __forceinline__ void async_wait() {
  asm volatile("s_wait_asynccnt 0" ::: "memory");
}

// Async global->LDS 128-bit copy (GV mode). lds_addr = low 32 bits of the
// generic shared-memory pointer (ISA: LDS aperture uses addr[31:0]).
__device__ __forceinline__ void async_g2l_b128(unsigned lds_addr, const void* gaddr) {
  asm volatile("global_load_async_to_lds_b128 %0, %1, off"
               :: "v"(lds_addr), "v"((unsigned long long)(size_t)gaddr)
               : "memory");
}

__global__ __launch_bounds__(THREADS) void qlstm_kernel(
    const float* __restrict__ x,      // [SEQ][BATCH][IN_DIM]
    const float* __restrict__ Wq,     // [32][256] (rows 0..7 used)
    const float* __restrict__ bq,     // [32]
    const float* __restrict__ pf, const float* __restrict__ pi_,
    const float* __restrict__ pg, const float* __restrict__ po,
    const float* __restrict__ Wf, const float* __restrict__ bf,
    const float* __restrict__ Wi, const float* __restrict__ bi,
    const float* __restrict__ Wg, const float* __restrict__ bg,
    const float* __restrict__ Wo, const float* __restrict__ bo,
    float* __restrict__ out)          // [SEQ*BATCH*HID] ++ hx ++ cx
{
  __shared__ float s_stRe[32*256];    // 32KB; overlays gOut[4][16][128] later
  __shared__ float s_stIm[32*256];    // 32KB
  __shared__ float s_comb[TILE_B*256];// [b][0:128]=x_t, [b][128:256]=h
  __shared__ float s_wq[16*256];      // rows 8..15 zero-padded (kills divergence)
  __shared__ float s_wt[4*128*8];     // Wf/Wi/Wg/Wo
  __shared__ float s_bq[8];
  __shared__ float s_bias[4*128];
  __shared__ float s_cst[TILE_B*128]; // cell state
  __shared__ float s_qin[TILE_B*8];
  __shared__ float s_part[8*256];     // GEMM1 per-wave partials
  __shared__ float s_ev[TILE_B*4*8];  // expectation values
  __shared__ float s_cr[64], s_sr[64];// RY cos/sin [gate][depth][wire]

  const int tid  = threadIdx.x;
  const int lane = tid & 31;
  const int wv   = tid >> 5;
  const int wgB  = blockIdx.x * TILE_B;

  // ---------------- one-time preload ----------------
  for (int i = tid; i < 16*256; i += THREADS) s_wq[i] = (i < 8*256) ? Wq[i] : 0.f;
  for (int i = tid; i < 1024; i += THREADS) {
    s_wt[0*1024+i] = Wf[i]; s_wt[1*1024+i] = Wi[i];
    s_wt[2*1024+i] = Wg[i]; s_wt[3*1024+i] = Wo[i];
  }
  if (tid < 8) s_bq[tid] = bq[tid];
  if (tid < 128) {
    s_bias[0*128+tid] = bf[tid]; s_bias[1*128+tid] = bi[tid];
    s_bias[2*128+tid] = bg[tid]; s_bias[3*128+tid] = bo[tid];
  }
  if (tid < 64) {
    int g = tid >> 4, rem = tid & 15;
    const float* P = (g==0) ? pf : (g==1) ? pi_ : (g==2) ? pg : po;
    float th = 0.5f * P[rem];
    s_cr[tid] = __cosf(th); s_sr[tid] = __sinf(th);
  }
  for (int i = tid; i < TILE_B*128; i += THREADS) {
    s_cst[i] = 0.f;
    int b = i >> 7, j = i & 127;
    s_comb[b*256 + 128 + j] = 0.f;  // h0 = 0
  }
  __syncthreads();

  // sim ownership: 8 lanes per sim, 4 sims per wave, 32 sims per pass
  const int lg = lane & 7;
  const int s  = wv*4 + (lane >> 3);
  const int sb = s & 15;            // batch row of this sim
  const int gateLocal = s >> 4;     // 0/1 within pass
  float* Re = s_stRe + s*256;
  float* Im = s_stIm + s*256;

  for (int t = 0; t < SEQ; ++t) {
    // ---- 1. async-copy x_t tile into combined[:, :128]; prefetch t+1 ----
    {
      const float4* xg = (const float4*)(x + ((size_t)t*BATCH + wgB)*IN_DIM);
      for (int q = tid; q < TILE_B*32; q += THREADS) {
        int b = q >> 5, k4 = q & 31;
        unsigned la = (unsigned)(size_t)&s_comb[b*256 + k4*4];
        async_g2l_b128(la, xg + b*32 + k4);
      }
      if (t + 1 < SEQ && tid < 64)  // locality 3 -> WGP-scope prefetch (all levels)
        __builtin_prefetch(x + ((size_t)(t+1)*BATCH + wgB)*IN_DIM + tid*32, 0, 3);
      async_wait();
    }
    __syncthreads();

    // ---- 2. GEMM1 via f32 WMMA: qin[16,8] = comb[16,256] @ Wq8^T ----
    {
      v8f acc = {0,0,0,0,0,0,0,0};
      const int row  = lane & 15;
      const int kOff = (lane >> 4) * 2;
      const int n    = lane & 15;             // cols 8..15 hit zero-padded rows
      const int kbase = wv * 32;              // K-split across 8 waves
      for (int kk = 0; kk < 8; ++kk) {
        int k0 = kbase + kk*4 + kOff;
        v2f a, bfr;
        a.x = s_comb[row*256 + k0];
        a.y = s_comb[row*256 + k0 + 1];
        bfr.x = s_wq[n*256 + k0];
        bfr.y = s_wq[n*256 + k0 + 1];
        acc = __builtin_amdgcn_wmma_f32_16x16x4_f32(
            false, a, false, bfr, (short)0, acc, false, false);
      }
      for (int r = 0; r < 8; ++r) s_part[wv*256 + r*32 + lane] = acc[r];
    }
    __syncthreads();
    {
      float sum = 0.f;
      for (int w = 0; w < 8; ++w) sum += s_part[w*256 + tid];
      int r = tid >> 5, l2 = tid & 31;
      int M = r + ((l2 >> 4) << 3), N = l2 & 15;
      if (N < 8) s_qin[M*8 + N] = sum + s_bq[N];
    }
    __syncthreads();

    // ---- 3. statevector sims: 2 passes x (2 gates x 16 batch) ----
    for (int pass = 0; pass < 2; ++pass) {
      const int gate = 2*pass + gateLocal;
      for (int q = 0; q < 32; ++q) { int i = lg*32 + q; Re[i] = 0.f; Im[i] = 0.f; }
      if (lg == 0) Re[0] = 1.f;
      lds_wave_sync();
      // RX encoding layer (angles from q_in)
      for (int w = 0; w < NQ; ++w) {
        int sh = 7 - w, stride = 1 << sh;
        float th = 0.5f * s_qin[sb*8 + w];
        float cc = __cosf(th), sn = __sinf(th);
        for (int pp = 0; pp < 16; ++pp) {
          int p  = lg*16 + pp;
          int i0 = ((p >> sh) << (sh+1)) | (p & (stride-1));
          int i1 = i0 | stride;
          float r0=Re[i0], m0=Im[i0], r1=Re[i1], m1=Im[i1];
          Re[i0] = cc*r0 + sn*m1;  Im[i0] = cc*m0 - sn*r1;
          Re[i1] = sn*m0 + cc*r1;  Im[i1] = cc*m1 - sn*r0;
        }
        lds_wave_sync();
      }
      // DEPTH x (RY layer + CNOT chain)
      for (int d = 0; d < 2; ++d) {
        for (int w = 0; w < NQ; ++w) {
          int sh = 7 - w, stride = 1 << sh;
          float cc = s_cr[gate*16 + d*8 + w], sn = s_sr[gate*16 + d*8 + w];
          for (int pp = 0; pp < 16; ++pp) {
            int p  = lg*16 + pp;
            int i0 = ((p >> sh) << (sh+1)) | (p & (stride-1));
            int i1 = i0 | stride;
            float r0=Re[i0], m0=Im[i0], r1=Re[i1], m1=Im[i1];
            Re[i0] = cc*r0 - sn*r1;  Im[i0] = cc*m0 - sn*m1;
            Re[i1] = sn*r0 + cc*r1;  Im[i1] = sn*m0 + cc*m1;
          }
          lds_wave_sync();
        }
        for (int w = 0; w < NQ-1; ++w) {
          int tb = 6 - w; // target bit; control = tb+1
          for (int jj = 0; jj < 8; ++jj) {
            int j = lg*8 + jj;
            int low  = j & ((1 << tb) - 1);
            int high = j >> tb;
            int ia = (high << (tb+2)) | (1 << (tb+1)) | low;
            int ib = ia | (1 << tb);
            float tr = Re[ia]; Re[ia] = Re[ib]; Re[ib] = tr;
            float ti = Im[ia]; Im[ia] = Im[ib]; Im[ib] = ti;
          }
          lds_wave_sync();
        }
      }
      // Z expectations, reduced across the 8 owning lanes by shfl_xor
      float acc[8] = {0,0,0,0,0,0,0,0};
      for (int q = 0; q < 32; ++q) {
        int i = lg*32 + q;
        float pr = Re[i]*Re[i] + Im[i]*Im[i];
        #pragma unroll
        for (int w = 0; w < 8; ++w) acc[w] += ((i >> (7-w)) & 1) ? -pr : pr;
      }
      #pragma unroll
      for (int off = 1; off < 8; off <<= 1) {
        #pragma unroll
        for (int w = 0; w < 8; ++w) acc[w] += __shfl_xor(acc[w], off, 32);
      }
      if (lg == 0)
        for (int w = 0; w < 8; ++w) s_ev[(sb*4 + gate)*8 + w] = acc[w];
      lds_wave_sync();
    }
    __syncthreads();

    // ---- 4. GEMM2 via f32 WMMA: gates[4][16][128] (overlay on s_stRe) ----
    float* gOut = s_stRe;
    {
      const int row  = lane & 15;
      const int kOff = (lane >> 4) * 2;
      const int nl   = lane & 15;
      for (int j4 = 0; j4 < 4; ++j4) {
        int job = wv*4 + j4;           // 32 jobs = 4 gates x 8 N-tiles
        int g = job >> 3, nt = job & 7;
        int n = nt*16 + nl;
        float bv = s_bias[g*128 + n];
        v8f acc = {bv,bv,bv,bv,bv,bv,bv,bv};
        for (int k0 = 0; k0 < 8; k0 += 4) {
          v2f a, bfr;
          a.x = s_ev[(row*4 + g)*8 + k0 + kOff];
          a.y = s_ev[(row*4 + g)*8 + k0 + kOff + 1];
          bfr.x = s_wt[(g*128 + n)*8 + k0 + kOff];
          bfr.y = s_wt[(g*128 + n)*8 + k0 + kOff + 1];
          acc = __builtin_amdgcn_wmma_f32_16x16x4_f32(
              false, a, false, bfr, (short)0, acc, false, false);
        }
        int Mhi = (lane >> 4) << 3;
        for (int r = 0; r < 8; ++r)
          gOut[(g*16 + (r + Mhi))*128 + n] = acc[r];
      }
    }
    __syncthreads();

    // ---- 5. pointwise LSTM update + stream h_t out ----
    {
      int e0 = tid * 8;
      int b = e0 >> 7, j0 = e0 & 127;
      float* outp = out + ((size_t)t*BATCH + wgB + b)*HID;
      for (int j = j0; j < j0 + 8; ++j) {
        float fv = gOut[(0*16 + b)*128 + j];
        float iv = gOut[(1*16 + b)*128 + j];
        float gv = gOut[(2*16 + b)*128 + j];
        float ov = gOut[(3*16 + b)*128 + j];
        float f  = 1.f / (1.f + __expf(-fv));
        float ii = 1.f / (1.f + __expf(-iv));
        float gg = tanhf(gv);
        float o  = 1.f / (1.f + __expf(-ov));
        float cn = f * s_cst[b*128 + j] + ii * gg;
        float hn = o * tanhf(cn);
        s_cst[b*128 + j] = cn;
        s_comb[b*256 + 128 + j] = hn;
        outp[j] = hn;
      }
    }
    __syncthreads();
  }

  // ---- final hx, cx ----
  {
    const size_t O1 = (size_t)SEQ*BATCH*HID;
    const size_t O2 = O1 + (size_t)BATCH*HID;
    int e0 = tid * 8;
    int b = e0 >> 7, j0 = e0 & 127;
    for (int j = j0; j < j0 + 8; ++j) {
      out[O1 + (size_t)(wgB + b)*HID + j] = s_comb[b*256 + 128 + j];
      out[O2 + (size_t)(wgB + b)*HID + j] = s_cst[b*128 + j];
    }
  }
}

extern "C" void kernel_launch(void* const* d_in, const int* in_sizes, int n_in,
                              void* d_out, int out_size, void* d_ws, size_t ws_size,
                              hipStream_t stream) {
  qlstm_kernel<<<BATCH/TILE_B, THREADS, 0, stream>>>(
      (const float*)d_in[0],  (const float*)d_in[1],  (const float*)d_in[2],
      (const float*)d_in[3],  (const float*)d_in[4],  (const float*)d_in[5],
      (const float*)d_in[6],  (const float*)d_in[7],  (const float*)d_in[8],
      (const float*)d_in[9],  (const float*)d_in[10], (const float*)d_in[11],
      (const float*)d_in[12], (const float*)d_in[13], (const float*)d_in[14],
      (float*)d_out);
}